// RandomWalkMobilityModel_39204461478191
// MI455X (gfx1250) — compile-verified
//
#include <hip/hip_runtime.h>
#include <math.h>

typedef __attribute__((ext_vector_type(2))) float v2f;
typedef __attribute__((ext_vector_type(8))) float v8f;

#define NR_   2048
#define ND_   2048
#define NCM_  32
#define BLOCK_ 256
#define WAVES_ (BLOCK_ / 32)            // 8 wave32 per block
#define DPT_   (ND_ / BLOCK_)           // 8 d-values per thread in scan phase
#define D_PER_WAVE_ (ND_ / WAVES_)      // 256 d-columns per wave in WMMA phase
#define TILES_PER_WAVE_ (D_PER_WAVE_ / 16) // 16 tiles of 16 columns

#define LOG2_F 0.6931471805599453f

__device__ __forceinline__ float softplus_f(float x) {
    // log(1 + e^x), stable for any x
    return fmaxf(x, 0.0f) + log1pf(expf(-fabsf(x)));
}

__global__ __launch_bounds__(BLOCK_) void rwm_fused_kernel(
    const float* __restrict__ CM_Alpha,
    const float* __restrict__ Wearing_Alpha,
    const float* __restrict__ Mobility_Alpha,
    const float* __restrict__ RegionR,
    const float* __restrict__ GI_mean,
    const float* __restrict__ GI_sd,
    const float* __restrict__ InitialSize_log,
    const float* __restrict__ noise,
    const float* __restrict__ psi,
    const float* __restrict__ NPIs,
    const float* __restrict__ ACMwear,
    const float* __restrict__ ACMmob,
    float* __restrict__ out)
{
    __shared__ float lds_gr[ND_];       // growth_reduction for this row (8 KB)
    __shared__ float lds_wsum[WAVES_];  // per-wave scan partials

    const int tid  = threadIdx.x;
    const int lane = tid & 31;
    const int wid  = tid >> 5;
    const int r    = blockIdx.x;

    const float* rowNPI = NPIs + (size_t)r * NCM_ * ND_;

    // ---------------- Phase 1: growth_reduction via V_WMMA_F32_16X16X4_F32 ---
    // A (16x4): A[m,k] = CM_Alpha[4i + k]  (identical across all 16 M-rows)
    // B (4x16): B[k,n] = NPI[r, 4i + k, d0 + n]
    // Lane->K mapping used identically for A and B fragments:
    //   .x -> K = (lane<16 ? 0 : 2),  .y -> K = (.x K) + 1
    // Result: every lane n (and n+16) holds out[d0 + (n&15)] in D[0].
    const int ksel = (lane < 16) ? 0 : 2;
    const int ncol = lane & 15;

    v2f afrag[NCM_ / 4];
    #pragma unroll
    for (int i = 0; i < NCM_ / 4; ++i) {
        afrag[i].x = CM_Alpha[4 * i + ksel];
        afrag[i].y = CM_Alpha[4 * i + ksel + 1];
    }

    for (int t = 0; t < TILES_PER_WAVE_; ++t) {
        const int d0 = wid * D_PER_WAVE_ + t * 16;
        const float* colp = rowNPI + d0 + ncol;
        v8f acc = {};
        #pragma unroll
        for (int i = 0; i < NCM_ / 4; ++i) {
            v2f b;
            b.x = colp[(size_t)(4 * i + ksel)     * ND_];
            b.y = colp[(size_t)(4 * i + ksel + 1) * ND_];
            acc = __builtin_amdgcn_wmma_f32_16x16x4_f32(
                false, afrag[i], false, b, (short)0, acc, false, false);
        }
        // All D rows identical; lanes l and l+16 write the same value.
        lds_gr[d0 + ncol] = acc[0];
    }
    __syncthreads();

    // ---------------- Phase 2: elementwise + row cumsum + output ------------
    const float wA   = Wearing_Alpha[0];
    const float mA   = Mobility_Alpha[0];
    const float gim  = GI_mean[0];
    const float gis  = GI_sd[0];
    const float gi_beta      = expf(gim / (gis * gis));
    const float inv_gi_alpha = (gis * gis) / (gim * gim);
    const float regR = RegionR[r];
    const float init = InitialSize_log[r];
    const float rr   = expf(psi[0]);
    // initial_mobility_reduction (d = 0 of this row)
    const float grm0 = softplus_f(mA * ACMmob[(size_t)r * ND_]) - LOG2_F;

    const size_t rowOff = (size_t)r * ND_ + (size_t)tid * DPT_;

    float4 w0 = *(const float4*)(ACMwear + rowOff);
    float4 w1 = *(const float4*)(ACMwear + rowOff + 4);
    float4 m0 = *(const float4*)(ACMmob  + rowOff);
    float4 m1 = *(const float4*)(ACMmob  + rowOff + 4);
    float4 n0 = *(const float4*)(noise   + rowOff);
    float4 n1 = *(const float4*)(noise   + rowOff + 4);

    float wv[DPT_] = {w0.x, w0.y, w0.z, w0.w, w1.x, w1.y, w1.z, w1.w};
    float mv[DPT_] = {m0.x, m0.y, m0.z, m0.w, m1.x, m1.y, m1.z, m1.w};
    float nv[DPT_] = {n0.x, n0.y, n0.z, n0.w, n1.x, n1.y, n1.z, n1.w};

    float vals[DPT_];
    float tot = 0.0f;
    #pragma unroll
    for (int i = 0; i < DPT_; ++i) {
        const float gr   = lds_gr[tid * DPT_ + i];
        const float wear = wA * wv[i];
        const float grm  = softplus_f(mA * mv[i]) - LOG2_F;
        const float elr  = regR - gr - wear - (grm - grm0);
        const float eg   = gi_beta + expf(elr) * inv_gi_alpha;
        vals[i] = eg + nv[i];
        tot += vals[i];
    }

    // inclusive wave32 scan of per-thread totals
    float sc = tot;
    #pragma unroll
    for (int off = 1; off < 32; off <<= 1) {
        float nbr = __shfl_up(sc, off, 32);
        if (lane >= off) sc += nbr;
    }
    if (lane == 31) lds_wsum[wid] = sc;
    __syncthreads();
    if (tid == 0) {
        float run = 0.0f;
        #pragma unroll
        for (int w = 0; w < WAVES_; ++w) {
            float v = lds_wsum[w];
            lds_wsum[w] = run;
            run += v;
        }
    }
    __syncthreads();

    // exclusive prefix for this thread over the whole row
    float run = init + lds_wsum[wid] + (sc - tot);
    #pragma unroll
    for (int i = 0; i < DPT_; ++i) {
        run += vals[i];                       // Log_Infected
        const float infected = expf(run);
        out[rowOff + i] = rr / (rr + infected) + 1e-8f;
    }
}

extern "C" void kernel_launch(void* const* d_in, const int* in_sizes, int n_in,
                              void* d_out, int out_size, void* d_ws, size_t ws_size,
                              hipStream_t stream) {
    (void)in_sizes; (void)n_in; (void)out_size; (void)d_ws; (void)ws_size;
    const float* CM_Alpha        = (const float*)d_in[0];
    const float* Wearing_Alpha   = (const float*)d_in[1];
    const float* Mobility_Alpha  = (const float*)d_in[2];
    const float* RegionR         = (const float*)d_in[3];
    const float* GI_mean         = (const float*)d_in[4];
    const float* GI_sd           = (const float*)d_in[5];
    const float* InitialSize_log = (const float*)d_in[6];
    const float* noise           = (const float*)d_in[7];
    const float* psi             = (const float*)d_in[8];
    const float* NPIs            = (const float*)d_in[9];
    const float* ACMwear         = (const float*)d_in[10];
    const float* ACMmob          = (const float*)d_in[11];
    float* out = (float*)d_out;

    rwm_fused_kernel<<<NR_, BLOCK_, 0, stream>>>(
        CM_Alpha, Wearing_Alpha, Mobility_Alpha, RegionR, GI_mean, GI_sd,
        InitialSize_log, noise, psi, NPIs, ACMwear, ACMmob, out);
}